// BGNNClassifier_42563125904013
// MI455X (gfx1250) — compile-verified
//
#include <hip/hip_runtime.h>
#include <hip/hip_bf16.h>

typedef __attribute__((ext_vector_type(2))) float v2f;
typedef __attribute__((ext_vector_type(8))) float v8f;
typedef int v4i_vs __attribute__((vector_size(16)));

#define FEAT 64

// ---------------- utility ----------------
__global__ void zero_f32_kernel(float* __restrict__ p, int n) {
  int i = blockIdx.x * blockDim.x + threadIdx.x;
  if (i < n) p[i] = 0.0f;
}

// ---------------- degree: cnt[dst] += 1 ----------------
__global__ void degree_kernel(const int* __restrict__ dst, float* __restrict__ cnt, int E) {
  int e = blockIdx.x * blockDim.x + threadIdx.x;
  if (e < E) unsafeAtomicAdd(&cnt[dst[e]], 1.0f);
}

// ---------------- scatter: summed[dst] += feat[src], one wave per edge ----------------
__global__ void scatter_kernel(const float* __restrict__ feat,
                               const int* __restrict__ src,
                               const int* __restrict__ dst,
                               float* __restrict__ summed, int E) {
  long long gid = (long long)blockIdx.x * blockDim.x + threadIdx.x;
  int e    = (int)(gid >> 5);   // 32 lanes (one wave) per edge
  int lane = (int)(gid & 31);   // lane handles 2 consecutive floats of the 64
  if (e >= E) return;
  int s = src[e];
  int d = dst[e];
  const float2 v = *reinterpret_cast<const float2*>(feat + (long long)s * FEAT + lane * 2);
  float* p = summed + (long long)d * FEAT + lane * 2;
  unsafeAtomicAdd(p,     v.x);
  unsafeAtomicAdd(p + 1, v.y);
}

// ---------------- fused SAGE node transform ----------------
// out = relu( (summed * 1/max(cnt,1)) @ Wl^T + bias + xin @ Wr^T )
// One wave handles a 16-node tile; 4 accumulators cover the 64 output cols.
// Weights (2 x 16KB) staged block-wide into LDS via async global->LDS loads
// (ASYNCcnt path), B operands then come from LDS (ds_load_b64). Matrix math
// on V_WMMA_F32_16X16X4_F32 (fp32 matrix core), K=64 -> 16 steps per matmul.
__global__ __launch_bounds__(256) void sage_gemm_kernel(
    const float* __restrict__ summed, const float* __restrict__ cnt,
    const float* __restrict__ xin,
    const float* __restrict__ Wl, const float* __restrict__ bias,
    const float* __restrict__ Wr,
    float* __restrict__ out, int n) {
  __shared__ float lds_w[2 * FEAT * FEAT];   // [0..4096) = Wl, [4096..8192) = Wr
  const int tid = threadIdx.x;

  // ---- block-wide weight staging (all threads participate, then barrier) ----
#if __has_builtin(__builtin_amdgcn_global_load_async_to_lds_b128)
  {
    typedef __attribute__((address_space(1))) v4i_vs gv4;
    typedef __attribute__((address_space(3))) v4i_vs lv4;
    gv4* gl = (gv4*)(Wl) + tid;                     // 16B units
    gv4* gr = (gv4*)(Wr) + tid;
    lv4* ll = (lv4*)(lds_w) + tid;
    lv4* lr = (lv4*)(lds_w + FEAT * FEAT) + tid;
    // 256 threads * 16B = 4KB per issue; 16KB per matrix = 4 issues each
    __builtin_amdgcn_global_load_async_to_lds_b128(gl, ll, 0,     0);
    __builtin_amdgcn_global_load_async_to_lds_b128(gl, ll, 4096,  0);
    __builtin_amdgcn_global_load_async_to_lds_b128(gl, ll, 8192,  0);
    __builtin_amdgcn_global_load_async_to_lds_b128(gl, ll, 12288, 0);
    __builtin_amdgcn_global_load_async_to_lds_b128(gr, lr, 0,     0);
    __builtin_amdgcn_global_load_async_to_lds_b128(gr, lr, 4096,  0);
    __builtin_amdgcn_global_load_async_to_lds_b128(gr, lr, 8192,  0);
    __builtin_amdgcn_global_load_async_to_lds_b128(gr, lr, 12288, 0);
#if __has_builtin(__builtin_amdgcn_s_wait_asynccnt)
    __builtin_amdgcn_s_wait_asynccnt(0);
#else
    asm volatile("s_wait_asynccnt 0" ::: "memory");
#endif
  }
#else
  for (int i = tid; i < FEAT * FEAT; i += 256) {
    lds_w[i]               = Wl[i];
    lds_w[FEAT * FEAT + i] = Wr[i];
  }
#endif
  __syncthreads();

  const int lane  = tid & 31;
  const int wave  = tid >> 5;
  const int tile  = blockIdx.x * 8 + wave;
  const int base  = tile * 16;
  if (base >= n) return;                  // wave-uniform: EXEC stays all-ones

  const int nlane = lane & 15;
  const int khalf = lane >> 4;            // 0 or 1 (k-offset group per ISA layout)
  int mrow = base + nlane;
  if (mrow >= n) mrow = n - 1;            // data clamp only, no divergence

  const float rcp = 1.0f / fmaxf(cnt[mrow], 1.0f);

  const float* arow = summed + (long long)mrow * FEAT + khalf * 2;
  const float* xrow = xin    + (long long)mrow * FEAT + khalf * 2;
  const float* wl   = lds_w + khalf * 2;                // B operand rows from LDS
  const float* wr   = lds_w + FEAT * FEAT + khalf * 2;

  v8f vz = {0.f, 0.f, 0.f, 0.f, 0.f, 0.f, 0.f, 0.f};
  v8f acc[4] = {vz, vz, vz, vz};

  // ---- aggr @ Wl^T  (mean fold-in via rcp) ----
  for (int kk = 0; kk < 16; ++kk) {
    const int kcol = kk * 4;
    float2 ta = *reinterpret_cast<const float2*>(arow + kcol);
    v2f a; a.x = ta.x * rcp; a.y = ta.y * rcp;
#pragma unroll
    for (int nt = 0; nt < 4; ++nt) {
      // B[k][ncol] = Wl[ncol][k]
      float2 tb = *reinterpret_cast<const float2*>(wl + (nt * 16 + nlane) * FEAT + kcol);
      v2f b; b.x = tb.x; b.y = tb.y;
      acc[nt] = __builtin_amdgcn_wmma_f32_16x16x4_f32(
          false, a, false, b, (short)0, acc[nt], false, false);
    }
  }

  // ---- x @ Wr^T ----
  for (int kk = 0; kk < 16; ++kk) {
    const int kcol = kk * 4;
    float2 ta = *reinterpret_cast<const float2*>(xrow + kcol);
    v2f a; a.x = ta.x; a.y = ta.y;
#pragma unroll
    for (int nt = 0; nt < 4; ++nt) {
      float2 tb = *reinterpret_cast<const float2*>(wr + (nt * 16 + nlane) * FEAT + kcol);
      v2f b; b.x = tb.x; b.y = tb.y;
      acc[nt] = __builtin_amdgcn_wmma_f32_16x16x4_f32(
          false, a, false, b, (short)0, acc[nt], false, false);
    }
  }

  // ---- bias + relu + store (C/D layout: m = r + 8*khalf, n = nt*16 + nlane) ----
#pragma unroll
  for (int nt = 0; nt < 4; ++nt) {
    const int ncol = nt * 16 + nlane;
    const float bv = bias[ncol];
#pragma unroll
    for (int r = 0; r < 8; ++r) {
      const int row = base + r + 8 * khalf;
      float v = acc[nt][r] + bv;
      v = fmaxf(v, 0.0f);
      if (row < n) out[(long long)row * FEAT + ncol] = v;
    }
  }
}

// ---------------- pooling: pooled[batch[node]] += h[node]; gcnt[g] += 1 ----------------
__global__ void pool_accum_kernel(const float* __restrict__ h, const int* __restrict__ batch,
                                  float* __restrict__ pooled, float* __restrict__ gcnt, int n) {
  long long idx = (long long)blockIdx.x * blockDim.x + threadIdx.x;
  if (idx >= (long long)n * FEAT) return;
  int node = (int)(idx >> 6);
  int f    = (int)(idx & 63);
  int g = batch[node];
  unsafeAtomicAdd(pooled + (long long)g * FEAT + f, h[idx]);
  if (f == 0) unsafeAtomicAdd(gcnt + g, 1.0f);
}

// ---------------- head: mean, FC, log_softmax; one block (64 thr) per graph ----------------
__global__ void head_kernel(const float* __restrict__ pooled, const float* __restrict__ gcnt,
                            const float* __restrict__ fcW, const float* __restrict__ fcb,
                            float* __restrict__ out) {
  const int g = blockIdx.x;
  const int t = threadIdx.x;      // 64 threads
  __shared__ float pg[64];
  __shared__ float logit[16];
  const float inv = 1.0f / fmaxf(gcnt[g], 1.0f);
  pg[t] = pooled[g * 64 + t] * inv;
  __syncthreads();
  if (t < 16) {
    float s = fcb[t];
    for (int f = 0; f < 64; ++f) s += pg[f] * fcW[t * 64 + f];
    logit[t] = s;
  }
  __syncthreads();
  if (t < 16) {
    float m = -1e30f;
    for (int o = 0; o < 16; ++o) m = fmaxf(m, logit[o]);
    float sum = 0.0f;
    for (int o = 0; o < 16; ++o) sum += __expf(logit[o] - m);
    out[g * 16 + t] = logit[t] - m - __logf(sum);
  }
}

extern "C" void kernel_launch(void* const* d_in, const int* in_sizes, int n_in,
                              void* d_out, int out_size, void* d_ws, size_t ws_size,
                              hipStream_t stream) {
  (void)n_in; (void)ws_size;
  const float* x   = (const float*)d_in[0];
  const float* W1l = (const float*)d_in[1];
  const float* b1  = (const float*)d_in[2];
  const float* W1r = (const float*)d_in[3];
  const float* W2l = (const float*)d_in[4];
  const float* b2  = (const float*)d_in[5];
  const float* W2r = (const float*)d_in[6];
  const float* fcW = (const float*)d_in[7];
  const float* fcb = (const float*)d_in[8];
  const int* edge  = (const int*)d_in[9];
  const int* batch = (const int*)d_in[10];

  const int N = in_sizes[10];
  const int E = in_sizes[9] / 2;
  const int G = out_size / 16;
  const int* srcp = edge;        // edge_index[0, :]
  const int* dstp = edge + E;    // edge_index[1, :]

  float* ws = (float*)d_ws;
  auto align64 = [](long long v) { return (v + 63) & ~63LL; };
  long long off = 0;
  float* cnt    = ws + off; off = align64(off + N);
  float* summed = ws + off; off = align64(off + (long long)N * FEAT);
  float* h1     = ws + off; off = align64(off + (long long)N * FEAT);
  float* h2     = ws + off; off = align64(off + (long long)N * FEAT);
  float* pooled = ws + off; off = align64(off + (long long)G * FEAT);
  float* gcnt   = ws + off; off = align64(off + G);

  const int nf      = N * FEAT;
  const int zblocks = (nf + 255) / 256;
  const int sblocks = (int)(((long long)E * 32 + 255) / 256);
  const int tiles   = (N + 15) / 16;
  const int gblocks = (tiles + 7) / 8;

  // degree (shared by both layers)
  zero_f32_kernel<<<(N + 255) / 256, 256, 0, stream>>>(cnt, N);
  degree_kernel<<<(E + 255) / 256, 256, 0, stream>>>(dstp, cnt, E);

  // layer 1
  zero_f32_kernel<<<zblocks, 256, 0, stream>>>(summed, nf);
  scatter_kernel<<<sblocks, 256, 0, stream>>>(x, srcp, dstp, summed, E);
  sage_gemm_kernel<<<gblocks, 256, 0, stream>>>(summed, cnt, x, W1l, b1, W1r, h1, N);

  // layer 2
  zero_f32_kernel<<<zblocks, 256, 0, stream>>>(summed, nf);
  scatter_kernel<<<sblocks, 256, 0, stream>>>(h1, srcp, dstp, summed, E);
  sage_gemm_kernel<<<gblocks, 256, 0, stream>>>(summed, cnt, h1, W2l, b2, W2r, h2, N);

  // pooling + head
  zero_f32_kernel<<<(G * FEAT + 255) / 256, 256, 0, stream>>>(pooled, G * FEAT);
  zero_f32_kernel<<<(G + 255) / 256, 256, 0, stream>>>(gcnt, G);
  pool_accum_kernel<<<zblocks, 256, 0, stream>>>(h2, batch, pooled, gcnt, N);
  head_kernel<<<G, 64, 0, stream>>>(pooled, gcnt, fcW, fcb, (float*)d_out);
}